// LSTM_53618371723874
// MI455X (gfx1250) — compile-verified
//
#include <hip/hip_runtime.h>

#define SEQ    512
#define NBATCH 256
#define NIN    150
#define NHID   4
#define NGATE  16          // 4*HID
#define NLAY   4
#define MROWS  (SEQ * NBATCH)   // 131072

#define WAVES_PER_BLK 4
#define TILE_FLTS (16 * NIN)    // 2400 floats / 9600 B per 16-row tile

typedef __attribute__((ext_vector_type(2))) float v2f;
typedef __attribute__((ext_vector_type(8))) float v8f;
typedef __attribute__((ext_vector_type(4))) unsigned int v4u;
typedef __attribute__((ext_vector_type(8))) int v8i;
typedef __attribute__((ext_vector_type(4))) int v4i;

// ---------------------------------------------------------------------------
// TDM: issue a 2-D tensor_load_to_lds of (rows x width) fp32, row stride
// `stride` elements, into LDS byte offset lds_off.  D# per ISA ch.8:
//  group0: [1:0]count=1 | lds_addr | global_addr lo | global_addr hi + type=2
//  group1: data_size=4B | tensor_dim0/1 | tile_dim0/1 | tensor_dim0_stride
// amdgpu-toolchain (clang-23) 6-arg builtin form:
//  (uint32x4 g0, int32x8 g1, int32x4 g2, int32x4 g3, int32x8 pad, i32 cpol)
// ---------------------------------------------------------------------------
static __device__ __forceinline__ void tdm_load_2d_f32(
    unsigned lds_off, const float* gptr, int width, int rows, int stride)
{
    const unsigned long long ga = (unsigned long long)(uintptr_t)gptr;
    v4u g0;
    g0[0] = 1u;                                        // count=1 (valid user D#)
    g0[1] = lds_off;                                   // LDS byte address
    g0[2] = (unsigned)(ga & 0xFFFFFFFFu);              // global addr [31:0]
    g0[3] = (unsigned)((ga >> 32) & 0x1FFFFFFu)        // global addr [56:32]
          | (2u << 30);                                // type = 2 ("image")
    v8i g1;
    g1[0] = (int)(2u << 16);                           // data_size = 4 bytes
    g1[1] = (int)((unsigned)(width & 0xFFFF) << 16);   // tensor_dim0[15:0]
    g1[2] = (int)(((unsigned)width >> 16)              // tensor_dim0[31:16]
          | ((unsigned)(rows & 0xFFFF) << 16));        // tensor_dim1[15:0]
    g1[3] = (int)(((unsigned)rows >> 16)               // tensor_dim1[31:16]
          | ((unsigned)(width & 0xFFFF) << 16));       // tile_dim0
    g1[4] = (int)(rows & 0xFFFF);                      // tile_dim1 (tile_dim2=0)
    g1[5] = (int)stride;                               // tensor_dim0_stride lo32
    g1[6] = 0;                                         // stride hi / dim1_stride lo
    g1[7] = 0;                                         // dim1_stride hi
    const v4i z4 = {0, 0, 0, 0};
    const v8i z8 = {0, 0, 0, 0, 0, 0, 0, 0};
    __builtin_amdgcn_tensor_load_to_lds(g0, g1, z4, z4, z8, 0);
}

// ---------------------------------------------------------------------------
// Kernel 1: layer-0 input projection with fp32 WMMA, TDM-staged operands.
//   xp[m][g] = sum_k x[m][k] * w[g][k] + (b_ih[g] + b_hh[g])
// One wave = one 16(M) x 16(N) tile.  x tile and w are DMA'd to LDS by the
// Tensor Data Mover; inner loop is ds_load_b64 fragments + WMMA only.
// A-layout (ISA 7.12.2, 32-bit A 16x4): lanes 0-15 -> K0,K1; lanes 16-31 ->
// K2,K3 (M = lane&15).  B mirrors with N = lane&15.
// C/D: acc[r] = D[M = r + 8*(lane>=16)][N = lane&15].
// ---------------------------------------------------------------------------
__global__ __launch_bounds__(32 * WAVES_PER_BLK) void lstm_l0_proj_wmma(
    const float* __restrict__ x,       // (MROWS, NIN)
    const float* __restrict__ w,       // (NGATE, NIN)
    const float* __restrict__ b_ih,    // (NGATE)
    const float* __restrict__ b_hh,    // (NGATE)
    float* __restrict__ xp)            // (MROWS, NGATE)
{
    __shared__ float smem[WAVES_PER_BLK * TILE_FLTS + TILE_FLTS];  // 48 KB
    const int WOFF = WAVES_PER_BLK * TILE_FLTS;                    // w at end

    const int lane = threadIdx.x & 31;
    const int wave = __builtin_amdgcn_readfirstlane((int)(threadIdx.x >> 5));
    const int tile = blockIdx.x * WAVES_PER_BLK + wave;
    const int rowBase = tile * 16;

    // ---- async DMA: this wave's 16x150 x-tile; wave 0 also DMAs w ----
    const unsigned xtile_off =
        (unsigned)(uintptr_t)(&smem[wave * TILE_FLTS]);   // flat->LDS offset
    tdm_load_2d_f32(xtile_off, x + (size_t)rowBase * NIN, NIN, 16, NIN);
    if (wave == 0) {
        const unsigned w_off = (unsigned)(uintptr_t)(&smem[WOFF]);
        tdm_load_2d_f32(w_off, w, NIN, NGATE, NIN);
    }
    __builtin_amdgcn_s_wait_tensorcnt(0);
    __syncthreads();                   // w visible to all waves

    const int mIdx = lane & 15;        // A row / B col within tile
    const int kOff = (lane >> 4) * 2;  // 0 for lanes 0-15, 2 for lanes 16-31

    const float* xrow = &smem[wave * TILE_FLTS + mIdx * NIN];  // LDS
    const float* wrow = &smem[WOFF + mIdx * NIN];              // LDS

    // bias folded into accumulator: every acc[r] of a lane is column mIdx
    const float bias = b_ih[mIdx] + b_hh[mIdx];
    v8f acc;
#pragma unroll
    for (int r = 0; r < 8; ++r) acc[r] = bias;

    // 37 full K-chunks: k 0..147, always in-bounds -> branch-free loop
#pragma unroll 4
    for (int k0 = 0; k0 < 148; k0 += 4) {
        const int ka = k0 + kOff;
        const float2 av = *(const float2*)(xrow + ka);   // ds_load_b64
        const float2 bv = *(const float2*)(wrow + ka);   // ds_load_b64
        v2f a = {av.x, av.y};
        v2f b = {bv.x, bv.y};
        acc = __builtin_amdgcn_wmma_f32_16x16x4_f32(
            false, a, false, b, (short)0, acc, false, false);
    }
    // tail chunk k=148: lanes 16-31 represent K=150,151 -> zero A (cndmask)
    {
        const float2 av = *(const float2*)(xrow + 148);  // in-bounds for all
        const float2 bv = *(const float2*)(wrow + 148);
        v2f a, b;
        a.x = kOff ? 0.0f : av.x;
        a.y = kOff ? 0.0f : av.y;
        b.x = bv.x;
        b.y = bv.y;
        acc = __builtin_amdgcn_wmma_f32_16x16x4_f32(
            false, a, false, b, (short)0, acc, false, false);
    }

    const int rOff = (lane >> 4) * 8;
#pragma unroll
    for (int r = 0; r < 8; ++r) {
        const int row = rowBase + rOff + r;
        xp[(size_t)row * NGATE + mIdx] = acc[r];
    }
}

// ---------------------------------------------------------------------------
// Kernel 2: fused 4-layer LSTM scan + FC, layer-pipelined wavefront.
// One wave (32 threads) per block: tid = layer*8 + localBatch.  Batch is
// independent -> 32 blocks x 8 batch spread over WGPs.  Single-wave
// workgroup: barrier is free, LDS handoff is in-order within the wave.
// Iteration k: layer l active for t = k - l in [0, SEQ).
// ---------------------------------------------------------------------------
__global__ __launch_bounds__(32) void lstm_scan_fused(
    const float* __restrict__ xp0,        // (SEQ, NBATCH, NGATE)
    const float* __restrict__ w_hh0,      // (NGATE, NHID)
    const float* __restrict__ w_ih_rest,  // (NLAY-1, NGATE, NHID)
    const float* __restrict__ w_hh_rest,  // (NLAY-1, NGATE, NHID)
    const float* __restrict__ b_ih_rest,  // (NLAY-1, NGATE)
    const float* __restrict__ b_hh_rest,  // (NLAY-1, NGATE)
    const float* __restrict__ fc_w,       // (1, NHID)
    const float* __restrict__ fc_b,       // (1,)
    float* __restrict__ out)              // (SEQ, NBATCH, 1)
{
    __shared__ float hbuf[2][NLAY - 1][8][NHID];   // 768 B

    const int tid   = threadIdx.x;
    const int layer = tid >> 3;                    // 0..3
    const int lb    = tid & 7;                     // local batch
    const int b     = blockIdx.x * 8 + lb;         // global batch

    float wih[NGATE][NHID];
    float whh[NGATE][NHID];
    float bias[NGATE];
    if (layer == 0) {
#pragma unroll
        for (int g = 0; g < NGATE; ++g) {
            bias[g] = 0.0f;
#pragma unroll
            for (int j = 0; j < NHID; ++j) {
                wih[g][j] = 0.0f;
                whh[g][j] = w_hh0[g * NHID + j];
            }
        }
    } else {
        const int l = layer - 1;
#pragma unroll
        for (int g = 0; g < NGATE; ++g) {
            bias[g] = b_ih_rest[l * NGATE + g] + b_hh_rest[l * NGATE + g];
#pragma unroll
            for (int j = 0; j < NHID; ++j) {
                wih[g][j] = w_ih_rest[(l * NGATE + g) * NHID + j];
                whh[g][j] = w_hh_rest[(l * NGATE + g) * NHID + j];
            }
        }
    }
    float fw[NHID];
#pragma unroll
    for (int j = 0; j < NHID; ++j) fw[j] = fc_w[j];
    const float fb = fc_b[0];

    float h[NHID] = {0.f, 0.f, 0.f, 0.f};
    float c[NHID] = {0.f, 0.f, 0.f, 0.f};

    for (int k = 0; k < SEQ + NLAY - 1; ++k) {
        const int  t      = k - layer;
        const bool active = (t >= 0) && (t < SEQ);
        if (active) {
            float g16[NGATE];
            if (layer == 0) {
                const float4* xpv =
                    (const float4*)(xp0 + ((size_t)t * NBATCH + b) * NGATE);
#pragma unroll
                for (int q = 0; q < 4; ++q) {
                    float4 v = xpv[q];
                    g16[q * 4 + 0] = v.x; g16[q * 4 + 1] = v.y;
                    g16[q * 4 + 2] = v.z; g16[q * 4 + 3] = v.w;
                }
            } else {
                float hin[NHID];
#pragma unroll
                for (int j = 0; j < NHID; ++j)
                    hin[j] = hbuf[(k - 1) & 1][layer - 1][lb][j];
#pragma unroll
                for (int g = 0; g < NGATE; ++g) {
                    float s = bias[g];
#pragma unroll
                    for (int j = 0; j < NHID; ++j) s += hin[j] * wih[g][j];
                    g16[g] = s;
                }
            }
#pragma unroll
            for (int g = 0; g < NGATE; ++g) {
                float s = g16[g];
#pragma unroll
                for (int j = 0; j < NHID; ++j) s += h[j] * whh[g][j];
                g16[g] = s;
            }
#pragma unroll
            for (int j = 0; j < NHID; ++j) {     // gate order i,f,g,o
                const float ig = 1.0f / (1.0f + __expf(-g16[j]));
                const float fg = 1.0f / (1.0f + __expf(-g16[NHID + j]));
                const float gg = tanhf(g16[2 * NHID + j]);
                const float og = 1.0f / (1.0f + __expf(-g16[3 * NHID + j]));
                c[j] = fg * c[j] + ig * gg;
                h[j] = og * tanhf(c[j]);
            }
            if (layer < NLAY - 1) {
#pragma unroll
                for (int j = 0; j < NHID; ++j)
                    hbuf[k & 1][layer][lb][j] = h[j];
            } else {
                float o = fb;
#pragma unroll
                for (int j = 0; j < NHID; ++j) o += h[j] * fw[j];
                out[(size_t)t * NBATCH + b] = o;
            }
        }
        __syncthreads();   // single-wave WG: lowers to waitcnt/NOP
    }
}

// ---------------------------------------------------------------------------
extern "C" void kernel_launch(void* const* d_in, const int* in_sizes, int n_in,
                              void* d_out, int out_size, void* d_ws, size_t ws_size,
                              hipStream_t stream) {
    (void)in_sizes; (void)n_in; (void)out_size; (void)ws_size;

    const float* x     = (const float*)d_in[0];
    const float* w_ih0 = (const float*)d_in[1];
    const float* w_hh0 = (const float*)d_in[2];
    const float* b_ih0 = (const float*)d_in[3];
    const float* b_hh0 = (const float*)d_in[4];
    const float* w_ihr = (const float*)d_in[5];
    const float* w_hhr = (const float*)d_in[6];
    const float* b_ihr = (const float*)d_in[7];
    const float* b_hhr = (const float*)d_in[8];
    const float* fc_w  = (const float*)d_in[9];
    const float* fc_b  = (const float*)d_in[10];

    float* out = (float*)d_out;
    float* xp0 = (float*)d_ws;                 // MROWS*NGATE fp32 = 8 MB

    // 8192 M-tiles of 16 rows, 4 waves (tiles) per 128-thread block
    lstm_l0_proj_wmma<<<MROWS / 16 / WAVES_PER_BLK, 32 * WAVES_PER_BLK, 0,
                        stream>>>(x, w_ih0, b_ih0, b_hh0, xp0);

    // 32 single-wave blocks, each: 8 batch x 4 pipelined layers
    lstm_scan_fused<<<NBATCH / 8, 32, 0, stream>>>(xp0, w_hh0, w_ihr, w_hhr,
                                                   b_ihr, b_hhr, fc_w, fc_b, out);
}